// Attention_2388001816931
// MI455X (gfx1250) — compile-verified
//
#include <hip/hip_runtime.h>
#include <math.h>

typedef __attribute__((ext_vector_type(2))) float v2f;
typedef __attribute__((ext_vector_type(8))) float v8f;

// D = A(16x4 f32) * B(4x16 f32) + C(16x16 f32)
#define WMMA4(a, b, c) \
  __builtin_amdgcn_wmma_f32_16x16x4_f32(false, (a), false, (b), (short)0, (c), false, false)

constexpr int BATCH = 4;
constexpr int SEQ   = 4096;
constexpr int DIMC  = 768;
constexpr int DC    = 48;               // head dim
constexpr float SCALE = 0.14433756729740643f;  // 1/sqrt(48)

// ---------------------------------------------------------------------------
// Kernel 1: q/k/v = x @ W + b.  256 threads = 8 waves; each wave owns a 16-row
// slab (128 rows/block). W chunks staged transposed in LDS -> B operand is one
// ds_load_b64 per WMMA. 9 accumulators (3 weights x 3 N-tiles of 16 = 48 cols).
// ---------------------------------------------------------------------------
__global__ __launch_bounds__(256) void qkv_kernel(
    const float* __restrict__ x,
    const float* __restrict__ Wq, const float* __restrict__ bq,
    const float* __restrict__ Wk, const float* __restrict__ bk,
    const float* __restrict__ Wv, const float* __restrict__ bv,
    float* __restrict__ qb, float* __restrict__ kb, float* __restrict__ vb)
{
  constexpr int KC = 64;                       // K-chunk staged in LDS
  __shared__ __align__(16) float wlds[3][DC * KC];   // [weight][n*KC + k] (transposed)

  const int t    = threadIdx.x;
  const int wave = t >> 5;
  const int lane = t & 31;
  const int col  = lane & 15;
  const int hi   = lane >> 4;

  const int blk     = blockIdx.x;              // BATCH * (SEQ/128) = 128 blocks
  const int b       = blk >> 5;                // 32 blocks per batch
  const int rowBase = (blk & 31) * 128 + wave * 16;

  const float* wsrc[3] = {Wq, Wk, Wv};
  const float* bias[3] = {bq, bk, bv};
  float*       outp[3] = {qb, kb, vb};

  v8f acc[3][3];
#pragma unroll
  for (int w = 0; w < 3; ++w)
#pragma unroll
    for (int nt = 0; nt < 3; ++nt) acc[w][nt] = {};

  const float* xrow = x + ((size_t)(b * SEQ + rowBase + col)) * DIMC;

  for (int kc = 0; kc < DIMC; kc += KC) {
    // cooperatively stage W chunk, transposed to [n][k]
#pragma unroll
    for (int w = 0; w < 3; ++w) {
      const float* Wg = wsrc[w] + (size_t)kc * DC;
      for (int e = t; e < DC * KC; e += 256) {
        int kk = e / DC;
        int n  = e - kk * DC;
        wlds[w][n * KC + kk] = Wg[(size_t)kk * DC + n];
      }
    }
    __syncthreads();

    for (int kk = 0; kk < KC; kk += 4) {
      v2f a = *(const v2f*)(xrow + kc + kk + 2 * hi);
#pragma unroll
      for (int w = 0; w < 3; ++w) {
#pragma unroll
        for (int nt = 0; nt < 3; ++nt) {
          v2f bm = *(const v2f*)(&wlds[w][(nt * 16 + col) * KC + kk + 2 * hi]);
          acc[w][nt] = WMMA4(a, bm, acc[w][nt]);
        }
      }
    }
    __syncthreads();
  }

  // C/D layout: VGPR r -> row r (lanes 0-15) / row r+8 (lanes 16-31), N = lane%16
#pragma unroll
  for (int w = 0; w < 3; ++w) {
#pragma unroll
    for (int nt = 0; nt < 3; ++nt) {
      float bia = bias[w][nt * 16 + col];
#pragma unroll
      for (int r = 0; r < 8; ++r) {
        int rowg = rowBase + r + 8 * hi;
        outp[w][((size_t)(b * SEQ + rowg)) * DC + nt * 16 + col] = acc[w][nt][r] + bia;
      }
    }
  }
}

// ---------------------------------------------------------------------------
// Kernel 2: per-(b, k-column) softmax stats over the QUERY axis.
// One wave per 16 k-columns; online max / sum-exp over all 256 q-tiles,
// recomputing S = scale * q k^T with WMMA (never materializing scores).
// ---------------------------------------------------------------------------
__global__ __launch_bounds__(32) void colstats_kernel(
    const float* __restrict__ qb, const float* __restrict__ kb,
    float* __restrict__ mOut, float* __restrict__ sOut)
{
  const int lane = threadIdx.x & 31;
  const int col  = lane & 15;
  const int hi   = lane >> 4;
  const int blk  = blockIdx.x;             // BATCH * (SEQ/16)
  const int b    = blk >> 8;
  const int kt   = blk & 255;

  // Preload k-tile as B operand: B[kk][n] = k[kt*16+n][kk] (D-contiguous rows)
  v2f kB[12];
  const float* krow = kb + ((size_t)(b * SEQ + kt * 16 + col)) * DC;
#pragma unroll
  for (int j = 0; j < 12; ++j) kB[j] = *(const v2f*)(krow + 4 * j + 2 * hi);

  float m = -INFINITY;
  float s = 0.0f;

  for (int qt = 0; qt < SEQ / 16; ++qt) {
    const float* qrow = qb + ((size_t)(b * SEQ + qt * 16 + col)) * DC;
    v8f c = {};
#pragma unroll
    for (int j = 0; j < 12; ++j) {
      v2f a = *(const v2f*)(qrow + 4 * j + 2 * hi);
      c = WMMA4(a, kB[j], c);
    }
    // column (N=col) reduction over 16 rows: 8 VGPRs + cross-half shuffle
    float tmax = -INFINITY;
#pragma unroll
    for (int r = 0; r < 8; ++r) tmax = fmaxf(tmax, c[r] * SCALE);
    tmax = fmaxf(tmax, __shfl_xor(tmax, 16));
    float mn = fmaxf(m, tmax);
    float tsum = 0.0f;
#pragma unroll
    for (int r = 0; r < 8; ++r) tsum += __expf(c[r] * SCALE - mn);
    tsum += __shfl_xor(tsum, 16);
    s = s * __expf(m - mn) + tsum;
    m = mn;
  }

  if (lane < 16) {
    mOut[(size_t)b * SEQ + kt * 16 + col] = m;
    sOut[(size_t)b * SEQ + kt * 16 + col] = s;
  }
}

// ---------------------------------------------------------------------------
// Kernel 3: out[q,:] = sum_k P[q,k] * v[k,:], P = exp(scale*qk - m_k)/s_k.
// One wave per 16 q-rows. S tile recomputed with WMMA; P bounced through LDS
// to convert C/D layout -> A layout for the P @ V WMMA.
// ---------------------------------------------------------------------------
__global__ __launch_bounds__(32) void attn_out_kernel(
    const float* __restrict__ qbuf, const float* __restrict__ kbuf,
    const float* __restrict__ vbuf,
    const float* __restrict__ mIn, const float* __restrict__ sIn,
    float* __restrict__ out)
{
  __shared__ __align__(16) float plds[16 * 16];

  const int lane = threadIdx.x & 31;
  const int col  = lane & 15;
  const int hi   = lane >> 4;
  const int blk  = blockIdx.x;             // BATCH * (SEQ/16)
  const int b    = blk >> 8;
  const int qt   = blk & 255;

  // Preload q-tile A operand (reused for all 256 k-tiles)
  v2f qA[12];
  const float* qrow = qbuf + ((size_t)(b * SEQ + qt * 16 + col)) * DC;
#pragma unroll
  for (int j = 0; j < 12; ++j) qA[j] = *(const v2f*)(qrow + 4 * j + 2 * hi);

  v8f acc[3];
  acc[0] = {}; acc[1] = {}; acc[2] = {};

  for (int kt = 0; kt < SEQ / 16; ++kt) {
    const float* krow = kbuf + ((size_t)(b * SEQ + kt * 16 + col)) * DC;
    v8f c = {};
#pragma unroll
    for (int j = 0; j < 12; ++j) {
      v2f kB = *(const v2f*)(krow + 4 * j + 2 * hi);
      c = WMMA4(qA[j], kB, c);
    }
    float mc = mIn[(size_t)b * SEQ + kt * 16 + col];
    float rs = 1.0f / sIn[(size_t)b * SEQ + kt * 16 + col];

    // P tile -> LDS (row-major), converting C/D layout to memory order
#pragma unroll
    for (int r = 0; r < 8; ++r)
      plds[(r + 8 * hi) * 16 + col] = __expf(c[r] * SCALE - mc) * rs;
    __syncthreads();

    // acc += P(16x16) @ V(16x48): A from LDS, B from v rows in global (L2-hot)
    const float* vtile = vbuf + ((size_t)(b * SEQ + kt * 16)) * DC;
#pragma unroll
    for (int ks = 0; ks < 16; ks += 4) {
      v2f a2 = *(const v2f*)(&plds[col * 16 + ks + 2 * hi]);
#pragma unroll
      for (int nt = 0; nt < 3; ++nt) {
        v2f b2;
        b2.x = vtile[(size_t)(ks + 2 * hi) * DC + nt * 16 + col];
        b2.y = vtile[(size_t)(ks + 2 * hi + 1) * DC + nt * 16 + col];
        acc[nt] = WMMA4(a2, b2, acc[nt]);
      }
    }
    __syncthreads();
  }

#pragma unroll
  for (int nt = 0; nt < 3; ++nt)
#pragma unroll
    for (int r = 0; r < 8; ++r)
      out[((size_t)(b * SEQ + qt * 16 + r + 8 * hi)) * DC + nt * 16 + col] = acc[nt][r];
}

// ---------------------------------------------------------------------------
extern "C" void kernel_launch(void* const* d_in, const int* in_sizes, int n_in,
                              void* d_out, int out_size, void* d_ws, size_t ws_size,
                              hipStream_t stream)
{
  const float* x  = (const float*)d_in[0];
  const float* Wq = (const float*)d_in[1];
  const float* bq = (const float*)d_in[2];
  const float* Wk = (const float*)d_in[3];
  const float* bk = (const float*)d_in[4];
  const float* Wv = (const float*)d_in[5];
  const float* bv = (const float*)d_in[6];
  float* out = (float*)d_out;

  const size_t qkvElems = (size_t)BATCH * SEQ * DC;   // 786432 floats each
  float* qb = (float*)d_ws;
  float* kb = qb + qkvElems;
  float* vb = kb + qkvElems;
  float* mC = vb + qkvElems;                          // [BATCH*SEQ]
  float* sC = mC + (size_t)BATCH * SEQ;               // [BATCH*SEQ]

  qkv_kernel<<<dim3(BATCH * (SEQ / 128)), dim3(256), 0, stream>>>(
      x, Wq, bq, Wk, bk, Wv, bv, qb, kb, vb);
  colstats_kernel<<<dim3(BATCH * (SEQ / 16)), dim3(32), 0, stream>>>(
      qb, kb, mC, sC);
  attn_out_kernel<<<dim3(BATCH * (SEQ / 16)), dim3(32), 0, stream>>>(
      qb, kb, vb, mC, sC, out);
}